// SegmentationDilation_52647709114949
// MI455X (gfx1250) — compile-verified
//
#include <hip/hip_runtime.h>
#include <stdint.h>

#define HDIM 512
#define WDIM 512
#define WPR 16                       // packed 32-bit words per row (global layout)
#define LSTRIDE 20                   // LDS words per row: rows 16B-aligned + bank-conflict-free
#define LWORDS (HDIM * LSTRIDE)      // 10240 words = 40 KB per LDS bitmap
#define RADIUS 10                    // FILTER_SIZE // 2
#define NPASSES 51                   // max(H,W) // (FILTER_SIZE//2)
#define FLOOD_THREADS 1024
#define ROWS_PT 8                    // consecutive rows owned per thread
#define HALO (ROWS_PT + 2 * RADIUS)  // 28 rows read per vertical window group

// d_ws layout:
//   bytes [0,8)        : u64 argmax key (zeroed each launch)
//   bytes [16, 16+32K) : packed foreground mask (HDIM*WPR u32), 16B aligned

typedef int v4i_t __attribute__((ext_vector_type(4)));
typedef __attribute__((address_space(1))) v4i_t* gv4i_p;   // global v4i*
typedef __attribute__((address_space(3))) v4i_t* lv4i_p;   // LDS v4i*

__device__ __forceinline__ uint32_t ord_f32(float f) {
    // monotonic total order on float bit patterns
    uint32_t u = __float_as_uint(f);
    return (u & 0x80000000u) ? ~u : (u | 0x80000000u);
}

__global__ void seg_init_kernel(unsigned long long* keySlot) {
    *keySlot = 0ull;
}

// mask bit = (in[...,1] - in[...,0]) > 0 (== round(sigmoid(diff)));
// argmax of diff with first-index tie-break via monotonic u64 key.
__global__ void seg_mask_argmax_kernel(const float* __restrict__ in,
                                       uint32_t* __restrict__ maskWords,
                                       unsigned long long* __restrict__ keySlot) {
    const int i = blockIdx.x * blockDim.x + threadIdx.x;   // pixel id, exact grid
    const float d = in[2 * i + 1] - in[2 * i];
    const bool fg = d > 0.0f;

#if __has_builtin(__builtin_amdgcn_ballot_w32)
    uint32_t m = __builtin_amdgcn_ballot_w32(fg);
    if ((threadIdx.x & 31) == 0) maskWords[i >> 5] = m;
#else
    if ((i & 31) == 0) maskWords[i >> 5] = 0u;
    __syncthreads();
    if (fg) atomicOr(&maskWords[i >> 5], 1u << (i & 31));
#endif

    __shared__ unsigned long long red[256];
    unsigned long long key =
        ((unsigned long long)ord_f32(d) << 32) | (uint32_t)(~(uint32_t)i);
    red[threadIdx.x] = key;
    __syncthreads();
    for (int s = 128; s > 0; s >>= 1) {
        if ((int)threadIdx.x < s) {
            unsigned long long o = red[threadIdx.x + s];
            if (o > red[threadIdx.x]) red[threadIdx.x] = o;
        }
        __syncthreads();
    }
    if (threadIdx.x == 0) atomicMax(keySlot, red[0]);
}

// Single-WGP LDS-resident bit-packed geodesic dilation (<= 51 passes, exact early exit).
__global__ void __launch_bounds__(FLOOD_THREADS)
seg_flood_kernel(const uint32_t* __restrict__ gmask,
                 const unsigned long long* __restrict__ keySlot,
                 float* __restrict__ out) {
    __shared__ __attribute__((aligned(16))) uint32_t sObj[LWORDS];  // current object bitmap
    __shared__ __attribute__((aligned(16))) uint32_t sTmp[LWORDS];  // staging / horiz result
    __shared__ int sChanged;

    const int tid = threadIdx.x;
    const int cw  = tid & (WPR - 1);        // column word 0..15
    const int rb  = (tid >> 4) * ROWS_PT;   // first owned row (0..504)

    // ---- stage packed mask into padded LDS via CDNA5 async global->LDS path ----
#if __has_builtin(__builtin_amdgcn_global_load_async_to_lds_b128) && \
    __has_builtin(__builtin_amdgcn_s_wait_asynccnt)
    {
        // 2048 b128 chunks: chunk c = row*4 + q  copies 4 words; both sides 16B aligned
        #pragma unroll
        for (int k = 0; k < 2; ++k) {
            const int c   = tid * 2 + k;
            const int row = c >> 2;
            const int q   = c & 3;
            __builtin_amdgcn_global_load_async_to_lds_b128(
                (gv4i_p)(gmask + row * WPR + q * 4),
                (lv4i_p)(sTmp + row * LSTRIDE + q * 4),
                0, 0);
        }
        __builtin_amdgcn_s_wait_asynccnt(0);   // this wave's ASYNCcnt -> 0
    }
#else
    for (int c = tid; c < HDIM * WPR; c += FLOOD_THREADS)
        sTmp[(c >> 4) * LSTRIDE + (c & 15)] = gmask[c];
#endif

    // zero object map (incl. pad words) while loads are in flight elsewhere
    for (int w = tid; w < LWORDS; w += FLOOD_THREADS) sObj[w] = 0u;
    if (tid == 0) sChanged = 0;
    __syncthreads();   // every wave waited on its own async loads before arriving

    // private mask words for the 8 owned rows
    uint32_t myMask[ROWS_PT];
    #pragma unroll
    for (int j = 0; j < ROWS_PT; ++j)
        myMask[j] = sTmp[(rb + j) * LSTRIDE + cw];
    __syncthreads();

    if (tid == 0) {
        const uint32_t seed = ~(uint32_t)(*keySlot);   // low 32 bits = ~flat_idx
        const int srow = seed >> 9;
        const int scol = seed & 511;
        sObj[srow * LSTRIDE + (scol >> 5)] = 1u << (scol & 31);
    }
    __syncthreads();

    for (int pass = 0; pass < NPASSES; ++pass) {
        uint32_t oldv[ROWS_PT];

        // ---- horizontal dilation radius 10: log-shift OR over 64-bit concat ----
        #pragma unroll
        for (int j = 0; j < ROWS_PT; ++j) {
            const int base = (rb + j) * LSTRIDE + cw;
            const uint32_t cur = sObj[base];
            const uint32_t L = (cw != 0)       ? sObj[base - 1] : 0u;
            const uint32_t R = (cw != WPR - 1) ? sObj[base + 1] : 0u;
            oldv[j] = cur;

            uint64_t u = ((uint64_t)cur << 32) | (uint64_t)L;   // shifts <<0..10
            uint64_t a = u | (u << 1);
            uint64_t b = a | (a << 2);
            uint64_t c = b | (b << 4);
            uint64_t dlo = c | ((a | (u << 2)) << 8);

            uint64_t v = ((uint64_t)R << 32) | (uint64_t)cur;   // shifts >>0..10
            uint64_t e = v | (v >> 1);
            uint64_t f = e | (e >> 2);
            uint64_t g = f | (f >> 4);
            uint64_t dhi = g | ((e | (v >> 2)) >> 8);

            sTmp[base] = (uint32_t)(dlo >> 32) | (uint32_t)dhi;
        }
        __syncthreads();

        // ---- vertical dilation radius 10 over 8 consecutive rows + mask ----
        uint32_t rowsv[HALO];
        #pragma unroll
        for (int d = 0; d < HALO; ++d) {
            const int rr = rb - RADIUS + d;
            rowsv[d] = (rr >= 0 && rr < HDIM) ? sTmp[rr * LSTRIDE + cw] : 0u;
        }
        // window j covers d in [j, j+20]; shared core d in [7, 20]
        uint32_t core = 0u;
        #pragma unroll
        for (int d = 7; d <= 20; ++d) core |= rowsv[d];
        uint32_t pre[7];
        pre[6] = rowsv[6];
        #pragma unroll
        for (int j = 5; j >= 0; --j) pre[j] = rowsv[j] | pre[j + 1];
        uint32_t post[8];
        post[1] = rowsv[21];
        #pragma unroll
        for (int j = 2; j <= 7; ++j) post[j] = post[j - 1] | rowsv[j + 20];

        int changed = 0;
        #pragma unroll
        for (int j = 0; j < ROWS_PT; ++j) {
            uint32_t v = core;
            if (j <= 6) v |= pre[j];
            if (j >= 1) v |= post[j];
            const uint32_t nv = v & myMask[j];
            changed |= (nv != oldv[j]);
            sObj[(rb + j) * LSTRIDE + cw] = nv;
        }
        if (changed) sChanged = 1;   // benign race: all writers store 1
        __syncthreads();

        const int ch = sChanged;     // uniform after barrier
        __syncthreads();             // all reads complete before reset
        if (tid == 0) sChanged = 0;
        if (!ch) break;              // fixpoint: remaining passes are identity
    }

    // ---- expand bits to float32 output [H, W, 1] ----
    for (int i = tid; i < HDIM * WDIM; i += FLOOD_THREADS) {
        const int row = i >> 9;
        const int col = i & 511;
        const uint32_t wv = sObj[row * LSTRIDE + (col >> 5)];
        out[i] = (float)((wv >> (col & 31)) & 1u);
    }
}

extern "C" void kernel_launch(void* const* d_in, const int* in_sizes, int n_in,
                              void* d_out, int out_size, void* d_ws, size_t ws_size,
                              hipStream_t stream) {
    (void)in_sizes; (void)n_in; (void)out_size; (void)ws_size;

    const float* in = (const float*)d_in[0];
    unsigned long long* keySlot = (unsigned long long*)d_ws;
    uint32_t* maskWords = (uint32_t*)((char*)d_ws + 16);
    float* out = (float*)d_out;

    seg_init_kernel<<<1, 1, 0, stream>>>(keySlot);
    seg_mask_argmax_kernel<<<(HDIM * WDIM) / 256, 256, 0, stream>>>(in, maskWords, keySlot);
    seg_flood_kernel<<<1, FLOOD_THREADS, 0, stream>>>(maskWords, keySlot, out);
}